// NFGI_30906584662714
// MI455X (gfx1250) — compile-verified
//
#include <hip/hip_runtime.h>
#include <hip/hip_bf16.h>

typedef __attribute__((ext_vector_type(2))) float v2f;
typedef __attribute__((ext_vector_type(8))) float v8f;

#define ROWS 8192
#define COLS 4096
#define CHUNK 256          // rows per pass2 block
#define COLS_PER_BLOCK 128 // 8 waves * 16 cols
#define HPAD 72            // padded per-wave histogram stride (>= 65), banks 8*w

// ---------------------------------------------------------------------------
// K0: initialize workspace accumulators (ws is poisoned; must re-init per call)
// ---------------------------------------------------------------------------
__global__ __launch_bounds__(256) void init_ws_kernel(float* __restrict__ colsum,
                                                      float* __restrict__ hist,
                                                      int* __restrict__ pminBits,
                                                      int* __restrict__ pmaxBits,
                                                      int n) {
  int i = blockIdx.x * blockDim.x + threadIdx.x;
  if (i < COLS) colsum[i] = 0.0f;
  if (i <= n)  hist[i] = 0.0f;
  if (i == 0) {
    *pminBits = 0x7f800000; // +inf
    *pmaxBits = 0;          // +0.0 (all p are positive)
  }
}

// ---------------------------------------------------------------------------
// K1: per-row stats. One block (256 thr) per row; each thread holds 16 floats.
// Produces rmis[r] = (rowMax, 1/sum(exp(x-max))) and the global p range:
//   max(p_row) = 1/S,  min(p_row) = exp(xmin - xmax)/S   (no p materialized)
// Positive floats order correctly as int bits -> atomicMin/Max on int.
// ---------------------------------------------------------------------------
__global__ __launch_bounds__(256) void row_stats_kernel(const float* __restrict__ seq,
                                                        float2* __restrict__ rmis,
                                                        int* __restrict__ pminBits,
                                                        int* __restrict__ pmaxBits) {
  const int r = blockIdx.x;
  const int t = threadIdx.x;
  const float4* row4 = (const float4*)(seq + (size_t)r * (size_t)COLS);

  float x[16];
#pragma unroll
  for (int i = 0; i < 4; ++i) {
    float4 v = row4[t * 4 + i];
    x[4 * i + 0] = v.x; x[4 * i + 1] = v.y;
    x[4 * i + 2] = v.z; x[4 * i + 3] = v.w;
  }

  float lmx = x[0], lmn = x[0];
#pragma unroll
  for (int i = 1; i < 16; ++i) {
    lmx = fmaxf(lmx, x[i]);
    lmn = fminf(lmn, x[i]);
  }

  __shared__ float red[256];
  __shared__ float bc[2];

  // block max
  red[t] = lmx; __syncthreads();
  for (int s = 128; s > 0; s >>= 1) {
    if (t < s) red[t] = fmaxf(red[t], red[t + s]);
    __syncthreads();
  }
  if (t == 0) bc[0] = red[0];
  __syncthreads();

  // block min
  red[t] = lmn; __syncthreads();
  for (int s = 128; s > 0; s >>= 1) {
    if (t < s) red[t] = fminf(red[t], red[t + s]);
    __syncthreads();
  }
  if (t == 0) bc[1] = red[0];
  __syncthreads();

  const float rmx = bc[0];
  float lsum = 0.0f;
#pragma unroll
  for (int i = 0; i < 16; ++i) lsum += __expf(x[i] - rmx); // arg <= 0, native-safe

  red[t] = lsum; __syncthreads();
  for (int s = 128; s > 0; s >>= 1) {
    if (t < s) red[t] += red[t + s];
    __syncthreads();
  }

  if (t == 0) {
    const float S  = red[0];
    const float is = 1.0f / S;
    rmis[r] = make_float2(rmx, is);
    atomicMax(pmaxBits, __float_as_int(is));                        // max p in row
    atomicMin(pminBits, __float_as_int(__expf(bc[1] - rmx) * is));  // min p in row
  }
}

// ---------------------------------------------------------------------------
// K2: streaming pass. Each wave owns a 16-column tile over a 256-row chunk.
// Per step it loads a 4x16 f32 tile into the WMMA B operand (lane layout:
// N = lane%16, K = 2*(lane/16)+v), accumulating column sums via
// V_WMMA_F32_16X16X4_F32 with A = ones(16x4). Row metadata for the lane's two
// rows (k0 even) comes from a single aligned b128 load of rmis[k0..k0+1].
// The same elements feed the softmax histogram through wave-private LDS bins,
// with per-lane run-length aggregation (softmax values are heavily
// bottom-skewed, so consecutive elements usually share a bin -> few
// ds_add_f32 instead of one per element).
// ---------------------------------------------------------------------------
__global__ __launch_bounds__(256) void pass2_kernel(const float* __restrict__ seq,
                                                    const float2* __restrict__ rmis,
                                                    const int* __restrict__ pminBits,
                                                    const int* __restrict__ pmaxBits,
                                                    float* __restrict__ colsum,
                                                    float* __restrict__ ghist,
                                                    int n) {
  const int wave = threadIdx.x >> 5;
  const int lane = threadIdx.x & 31;
  const int lhalf = lane >> 4;   // 0: lanes 0-15, 1: lanes 16-31
  const int ncol  = lane & 15;

  const int tiles_c  = COLS / COLS_PER_BLOCK;           // 32
  const int cb       = blockIdx.x % tiles_c;
  const int rc       = blockIdx.x / tiles_c;
  const int colBase  = cb * COLS_PER_BLOCK + wave * 16;
  const int col      = colBase + ncol;
  const int r0       = rc * CHUNK;

  __shared__ float lhist[8][HPAD];
  for (int i = threadIdx.x; i < 8 * HPAD; i += blockDim.x)
    ((float*)lhist)[i] = 0.0f;
  __syncthreads();

  const float mn   = __int_as_float(*pminBits);
  const float mx   = __int_as_float(*pmaxBits);
  const float invd = (float)n / (mx - mn);

  v2f a; a[0] = 1.0f; a[1] = 1.0f;  // ones 16x4 A-matrix
  v8f c = {};

  int   curBin = -1;   // per-lane histogram run-length aggregation
  float acc    = 0.0f;

  const float4* rmis4 = (const float4*)rmis; // (rm0, is0, rm1, is1) per pair

#pragma unroll 2
  for (int rr = 0; rr < CHUNK; rr += 4) {
    const int k0 = r0 + rr + 2 * lhalf;          // even -> 16B-aligned pair
    const float4 m = rmis4[k0 >> 1];             // rows k0 and k0+1 metadata

    v2f b;
    const float xv0 = seq[(size_t)k0 * (size_t)COLS + col];
    const float xv1 = seq[(size_t)(k0 + 1) * (size_t)COLS + col];
    b[0] = xv0;
    b[1] = xv1;

#pragma unroll
    for (int v = 0; v < 2; ++v) {
      const float xv = v ? xv1 : xv0;
      const float rm = v ? m.z : m.x;
      const float is = v ? m.w : m.y;

      // softmax value and histogram bin (bin n == overflow, dropped later)
      const float p = __expf(xv - rm) * is;
      int bin = (int)((p - mn) * invd);
      bin = bin < 0 ? 0 : (bin > n ? n : bin);
      if (bin == curBin) {
        acc += p;
      } else {
        if (curBin >= 0) atomicAdd(&lhist[wave][curBin], acc);
        curBin = bin;
        acc = p;
      }
    }

    // prefetch a future row segment of this column tile (global_prefetch_b8)
    if ((rr & 31) == 0) {
      int pk = r0 + rr + 64;
      if (pk > ROWS - 1) pk = ROWS - 1;
      __builtin_prefetch(seq + (size_t)pk * (size_t)COLS + col, 0, 1);
    }

    // D[m][n] = sum_k B[k][n] + C[m][n]  -> every row of D = running col sums
    c = __builtin_amdgcn_wmma_f32_16x16x4_f32(false, a, false, b,
                                              (short)0, c, false, false);
  }
  if (curBin >= 0) atomicAdd(&lhist[wave][curBin], acc); // flush tail run

  // C/D layout: VGPR0, lanes 0-15 hold (M=0, N=lane) -> the 16 column sums.
  if (lane < 16) atomicAdd(&colsum[colBase + lane], c[0]);

  __syncthreads();
  for (int bin = threadIdx.x; bin <= n; bin += blockDim.x) {
    float s = 0.0f;
#pragma unroll
    for (int w = 0; w < 8; ++w) s += lhist[w][bin];
    atomicAdd(&ghist[bin], s);
  }
}

// ---------------------------------------------------------------------------
// K3: write output: [0,COLS) = column means, [COLS, COLS+n) = histogram bins
// ---------------------------------------------------------------------------
__global__ __launch_bounds__(256) void finalize_kernel(const float* __restrict__ colsum,
                                                       const float* __restrict__ hist,
                                                       float* __restrict__ out,
                                                       int n) {
  int i = blockIdx.x * blockDim.x + threadIdx.x;
  if (i < COLS) {
    out[i] = colsum[i] * (1.0f / (float)ROWS);
  } else if (i < COLS + n) {
    out[i] = hist[i - COLS];
  }
}

// ---------------------------------------------------------------------------
extern "C" void kernel_launch(void* const* d_in, const int* in_sizes, int n_in,
                              void* d_out, int out_size, void* d_ws, size_t ws_size,
                              hipStream_t stream) {
  const float* seq = (const float*)d_in[0];
  const int n = out_size - COLS; // 64 per reference

  // workspace layout (floats); rmis rounded up to a 16B boundary so the
  // b128 pair-loads in pass2 are naturally aligned (hipMalloc is 256B-aligned)
  float* ws       = (float*)d_ws;
  float* colsum   = ws;                        // COLS
  float* hist     = ws + COLS;                 // n+1
  int*   pminBits = (int*)(ws + COLS + (n + 1));
  int*   pmaxBits = pminBits + 1;
  size_t rmisOff  = (size_t)(COLS + (n + 1) + 2);
  rmisOff         = (rmisOff + 3) & ~(size_t)3; // 16B-aligned float index
  float2* rmis    = (float2*)(ws + rmisOff);    // ROWS float2

  // K0: init accumulators
  {
    int threads = 256;
    int blocks = (COLS + threads - 1) / threads;
    init_ws_kernel<<<blocks, threads, 0, stream>>>(colsum, hist, pminBits, pmaxBits, n);
  }

  // K1: per-row max / min / sum-exp  +  global p-range
  row_stats_kernel<<<ROWS, 256, 0, stream>>>(seq, rmis, pminBits, pmaxBits);

  // K2: WMMA column sums + softmax histogram
  {
    int blocks = (COLS / COLS_PER_BLOCK) * (ROWS / CHUNK); // 32 * 32 = 1024
    pass2_kernel<<<blocks, 256, 0, stream>>>(seq, rmis, pminBits, pmaxBits,
                                             colsum, hist, n);
  }

  // K3: emit [mean | hist]
  {
    int threads = 256;
    int blocks = (COLS + n + threads - 1) / threads;
    finalize_kernel<<<blocks, threads, 0, stream>>>(colsum, hist, (float*)d_out, n);
  }
}